// MultiHeadClassifier_47278999994542
// MI455X (gfx1250) — compile-verified
//
#include <hip/hip_runtime.h>
#include <math.h>

typedef float v2f __attribute__((ext_vector_type(2)));
typedef float v8f __attribute__((ext_vector_type(8)));

#define N_PTS   32768
#define IN_F    256
#define NUM_CAT 16
#define CF      4096          // NUM_CAT * IN_F
#define SEG     6
#define OUT_K   50
#define BN_EPS  1e-5f
#define MAX_SUPER 528         // 32768/64 + 16 (per-category padding to 64 rows)
#define PADDED_MAX 33792      // MAX_SUPER * 64

// ---- workspace layout (in 32-bit words) ----
#define OFF_G       0         // 256*256 f32 Gram
#define OFF_S       65536     // 256 f32 column sums of features
#define OFF_COUNTS  65792     // 16 i32
#define OFF_CURSORS 65808     // 16 i32
#define OFF_OFFC    65824     // 16 i32 padded category start positions
#define OFF_NT      65840     // 1 i32 (padded to 16)
#define OFF_TILECAT 65856     // 544 i32 (super-tile -> category)
#define OFF_SCALE   66400     // 4096 f32
#define OFF_SHIFT   70496     // 4096 f32
#define OFF_PERM    74592     // 33792 i32
#define OFF_Z       108416    // 33792*256 f32 (~34.6 MB)

__global__ void k_zero(float* p, int n) {
  int i = blockIdx.x * 256 + threadIdx.x;
  if (i < n) p[i] = 0.0f;
}

// column sums of features -> s[256]
__global__ void k_colsum(const float* __restrict__ feat, float* __restrict__ s) {
  int f = threadIdx.x;                   // 256 threads
  int r0 = blockIdx.x * 256;             // 128 blocks * 256 rows
  float acc = 0.0f;
  for (int r = 0; r < 256; ++r) acc += feat[(size_t)(r0 + r) * IN_F + f];
  atomicAdd(&s[f], acc);
}

// Gram G = feat^T * feat (upper-triangle tiles) via V_WMMA_F32_16X16X4_F32.
// Block = 16 waves; wave tj computes tile (ti, tj) if tj >= ti.
// A slab (16 cols of features x 64 K) staged in LDS, shared by all waves.
__global__ void __launch_bounds__(512) k_gram(const float* __restrict__ feat,
                                              float* __restrict__ G) {
  __shared__ float Alds[16][68];         // stride 68 -> conflict-free reads
  int ti = blockIdx.x;                   // 0..15
  int kc = blockIdx.y;                   // 0..15, chunk of 2048 rows
  int tid = threadIdx.x;
  int wave = tid >> 5, lane = tid & 31;
  int half = lane >> 4, l15 = lane & 15;
  int tj = wave;
  int i0 = ti * 16, j0 = tj * 16;
  bool active = (tj >= ti);              // symmetric: skip lower-triangle tiles
  v8f acc = {};
  for (int ks = 0; ks < 2048; ks += 64) {
    int kbase = kc * 2048 + ks;
    __syncthreads();
    // load A slab: Alds[m][kk] = feat[kbase+kk][i0+m]  (1024 floats, 2/thread)
    for (int i = 0; i < 2; ++i) {
      int idx = tid + i * 512;
      int kk = idx >> 4, m = idx & 15;
      Alds[m][kk] = feat[(size_t)(kbase + kk) * IN_F + i0 + m];
    }
    __syncthreads();
    if (active) {
      for (int k = 0; k < 64; k += 4) {
        int K = k + half * 2;            // A/B lane layout per ISA 7.12.2
        v2f a = *(const v2f*)&Alds[l15][K];
        v2f b;
        b.x = feat[(size_t)(kbase + K) * IN_F + j0 + l15];
        b.y = feat[(size_t)(kbase + K + 1) * IN_F + j0 + l15];
        acc = __builtin_amdgcn_wmma_f32_16x16x4_f32(false, a, false, b,
                                                    (short)0, acc, false, false);
      }
    }
  }
  if (active) {
    for (int r = 0; r < 8; ++r) {
      int m = r + half * 8;
      atomicAdd(&G[(size_t)(i0 + m) * IN_F + j0 + l15], acc[r]);
    }
  }
}

// fill lower triangle from upper
__global__ void k_mirror(float* __restrict__ G) {
  int idx = blockIdx.x * 256 + threadIdx.x;   // 65536
  int i = idx >> 8, j = idx & 255;
  if ((i >> 4) > (j >> 4)) G[(size_t)i * IN_F + j] = G[(size_t)j * IN_F + i];
}

__global__ void k_hist(const int* __restrict__ cats, int* __restrict__ counts) {
  int n = blockIdx.x * 256 + threadIdx.x;
  if (n < N_PTS) atomicAdd(&counts[cats[n]], 1);
}

// planning: category offsets padded to 64 rows + super-tile -> category map
__global__ void k_plan(const int* __restrict__ counts, int* __restrict__ offc,
                       int* __restrict__ tileCat, int* __restrict__ nT) {
  int pos = 0, t = 0;
  for (int c = 0; c < NUM_CAT; ++c) {
    offc[c] = pos;
    int nt = (counts[c] + 63) >> 6;
    for (int i = 0; i < nt; ++i) tileCat[t++] = c;
    pos += nt * 64;
  }
  nT[0] = t;
  for (int i = t; i < MAX_SUPER; ++i) tileCat[i] = -1;
}

__global__ void k_scatter(const int* __restrict__ cats, const int* __restrict__ offc,
                          int* __restrict__ cursors, int* __restrict__ perm) {
  int n = blockIdx.x * 256 + threadIdx.x;
  if (n < N_PTS) {
    int c = cats[n];
    int idx = atomicAdd(&cursors[c], 1);
    perm[offc[c] + idx] = n;
  }
}

__global__ void k_padfill(const int* __restrict__ counts, const int* __restrict__ offc,
                          int* __restrict__ perm) {
  int c = blockIdx.x;                     // 16 blocks, 64 threads
  int cnt = counts[c];
  int padded = ((cnt + 63) >> 6) << 6;
  int fill = (cnt > 0) ? perm[offc[c]] : 0;
  for (int i = cnt + threadIdx.x; i < padded; i += blockDim.x) perm[offc[c] + i] = fill;
}

// per output column j: mu = (s.w)/N ; E[x^2] = (w^T G w)/N -> scale/shift
__global__ void k_stats(const float* __restrict__ W1, const float* __restrict__ G,
                        const float* __restrict__ s, const float* __restrict__ gamma,
                        const float* __restrict__ beta,
                        float* __restrict__ scale, float* __restrict__ shift) {
  __shared__ float w[IN_F];
  __shared__ float red[IN_F];
  __shared__ float red2[IN_F];
  int j = blockIdx.x;                     // 4096 blocks
  int f = threadIdx.x;                    // 256 threads
  float wf = W1[(size_t)f * CF + j];
  w[f] = wf;
  __syncthreads();
  float t = 0.0f;
  for (int k = 0; k < IN_F; ++k) t += G[(size_t)f * IN_F + k] * w[k];
  red[f]  = wf * t;
  red2[f] = s[f] * wf;
  __syncthreads();
  for (int st = 128; st > 0; st >>= 1) {
    if (f < st) { red[f] += red[f + st]; red2[f] += red2[f + st]; }
    __syncthreads();
  }
  if (f == 0) {
    const float invN = 1.0f / (float)N_PTS;
    float mu  = red2[0] * invN;
    float m2  = red[0] * invN;
    float var = m2 - mu * mu;
    float sc  = gamma[j] * rsqrtf(var + BN_EPS);
    scale[j] = sc;
    shift[j] = beta[j] - mu * sc;
  }
}

// bucketed GEMM, register-blocked: block = 4 row tiles x 16 col tiles of one
// category; A (64x64 K-slab) via LDS shared by 16 waves; each B load feeds
// 4 WMMAs (4 independent accumulator chains).
__global__ void __launch_bounds__(512) k_gemm2(
    const float* __restrict__ feat, const float* __restrict__ W1,
    const float* __restrict__ scale, const float* __restrict__ shift,
    const int* __restrict__ perm, const int* __restrict__ tileCat,
    const int* __restrict__ offc, const int* __restrict__ counts,
    float* __restrict__ z) {
  __shared__ float Alds[64][68];
  __shared__ int rows[64];
  int sb = blockIdx.x;                    // super tile (64 rows)
  int c = tileCat[sb];
  if (c < 0) return;                      // block-uniform
  int tid = threadIdx.x;
  int wave = tid >> 5, lane = tid & 31;
  int half = lane >> 4, l15 = lane & 15;
  int row0 = sb * 64;
  if (tid < 64) rows[tid] = perm[row0 + tid];
  int j0 = wave * 16;
  int colbase = c * IN_F + j0;
  v8f accs[4] = {};
  for (int ks = 0; ks < IN_F; ks += 64) {
    __syncthreads();
    // load 64x64 A slab (4096 floats, 8/thread, fully coalesced)
    for (int i = 0; i < 8; ++i) {
      int idx = tid + i * 512;
      int m = idx >> 6, kk = idx & 63;
      Alds[m][kk] = feat[(size_t)rows[m] * IN_F + ks + kk];
    }
    __syncthreads();
    if (ks + 64 < IN_F) {                 // prefetch next W1 K-slab
      __builtin_prefetch(&W1[(size_t)(ks + 64 + half * 2) * CF + colbase + l15], 0, 1);
    }
    for (int k = 0; k < 64; k += 4) {
      int K = k + half * 2;
      int gk = ks + K;
      v2f b;
      b.x = W1[(size_t)gk * CF + colbase + l15];
      b.y = W1[(size_t)(gk + 1) * CF + colbase + l15];
      v2f a0 = *(const v2f*)&Alds[l15][K];
      v2f a1 = *(const v2f*)&Alds[16 + l15][K];
      v2f a2 = *(const v2f*)&Alds[32 + l15][K];
      v2f a3 = *(const v2f*)&Alds[48 + l15][K];
      accs[0] = __builtin_amdgcn_wmma_f32_16x16x4_f32(false, a0, false, b,
                                                      (short)0, accs[0], false, false);
      accs[1] = __builtin_amdgcn_wmma_f32_16x16x4_f32(false, a1, false, b,
                                                      (short)0, accs[1], false, false);
      accs[2] = __builtin_amdgcn_wmma_f32_16x16x4_f32(false, a2, false, b,
                                                      (short)0, accs[2], false, false);
      accs[3] = __builtin_amdgcn_wmma_f32_16x16x4_f32(false, a3, false, b,
                                                      (short)0, accs[3], false, false);
    }
  }
  float sc = scale[colbase + l15];
  float sh = shift[colbase + l15];
  int cnt = counts[c];
  int oc  = offc[c];
#pragma unroll
  for (int rt = 0; rt < 4; ++rt) {
#pragma unroll
    for (int r = 0; r < 8; ++r) {
      int m = rt * 16 + r + half * 8;
      int pos = row0 + m;
      if (pos - oc < cnt) {
        float v = accs[rt][r] * sc + sh;
        v = (v >= 0.0f) ? v : 0.2f * v;   // LeakyReLU(0.2)
        z[(size_t)pos * IN_F + j0 + l15] = v;
      }
    }
  }
}

// wave-per-point: logits = z . Wc[c] + bias ; log_softmax over 6 ; scatter [N,50]
__global__ void k_final(const float* __restrict__ z, const float* __restrict__ Wc,
                        const float* __restrict__ bias,
                        const int* __restrict__ perm, const int* __restrict__ tileCat,
                        const int* __restrict__ offc, const int* __restrict__ counts,
                        const int* __restrict__ shifts, const int* __restrict__ seg_lens,
                        float* __restrict__ out) {
  int wave = threadIdx.x >> 5;            // 8 waves/block (wave32)
  int lane = threadIdx.x & 31;
  int pos = blockIdx.x * 8 + wave;
  int t = pos >> 6;                       // super tile
  int c = tileCat[t];
  if (c < 0) return;
  if (pos - offc[c] >= counts[c]) return; // padding slot
  int n = perm[pos];
  float a0 = 0, a1 = 0, a2 = 0, a3 = 0, a4 = 0, a5 = 0;
  const float* wc = Wc + (size_t)c * IN_F * SEG;
  const float* zr = z + (size_t)pos * IN_F;
  for (int f = lane; f < IN_F; f += 32) {
    float zv = zr[f];
    const float* w = wc + f * SEG;
    a0 += zv * w[0]; a1 += zv * w[1]; a2 += zv * w[2];
    a3 += zv * w[3]; a4 += zv * w[4]; a5 += zv * w[5];
  }
  for (int m = 16; m >= 1; m >>= 1) {
    a0 += __shfl_xor(a0, m, 32); a1 += __shfl_xor(a1, m, 32);
    a2 += __shfl_xor(a2, m, 32); a3 += __shfl_xor(a3, m, 32);
    a4 += __shfl_xor(a4, m, 32); a5 += __shfl_xor(a5, m, 32);
  }
  float l[SEG];
  l[0] = a0 + bias[0]; l[1] = a1 + bias[1]; l[2] = a2 + bias[2];
  l[3] = a3 + bias[3]; l[4] = a4 + bias[4]; l[5] = a5 + bias[5];
  float mx = l[0];
  for (int sI = 1; sI < SEG; ++sI) mx = fmaxf(mx, l[sI]);
  float se = 0.0f;
  for (int sI = 0; sI < SEG; ++sI) se += __expf(l[sI] - mx);
  float lz = __logf(se) + mx;
  float lsm[SEG];
  for (int sI = 0; sI < SEG; ++sI) lsm[sI] = l[sI] - lz;
  int sh = shifts[c];
  int len = seg_lens[c];
  float* orow = out + (size_t)n * OUT_K;
  for (int k = lane; k < OUT_K; k += 32) {
    int jj = k - sh;
    orow[k] = (jj >= 0 && jj < len) ? lsm[jj] : 0.0f;
  }
}

extern "C" void kernel_launch(void* const* d_in, const int* in_sizes, int n_in,
                              void* d_out, int out_size, void* d_ws, size_t ws_size,
                              hipStream_t stream) {
  const float* feat   = (const float*)d_in[0];
  const float* W1     = (const float*)d_in[1];
  const float* gamma  = (const float*)d_in[2];
  const float* beta   = (const float*)d_in[3];
  const float* Wc     = (const float*)d_in[4];
  const float* bias   = (const float*)d_in[5];
  const int*   cats   = (const int*)d_in[6];
  const int*   shifts = (const int*)d_in[7];
  const int*   slens  = (const int*)d_in[8];
  float* out = (float*)d_out;
  float* ws  = (float*)d_ws;

  float* G      = ws + OFF_G;
  float* s      = ws + OFF_S;
  int*   counts = (int*)(ws + OFF_COUNTS);
  int*   cursors= (int*)(ws + OFF_CURSORS);
  int*   offc   = (int*)(ws + OFF_OFFC);
  int*   nT     = (int*)(ws + OFF_NT);
  int*   tileCat= (int*)(ws + OFF_TILECAT);
  float* scale  = ws + OFF_SCALE;
  float* shift  = ws + OFF_SHIFT;
  int*   perm   = (int*)(ws + OFF_PERM);
  float* z      = ws + OFF_Z;

  // zero G, s, counts, cursors (contiguous 65824 words)
  k_zero<<<(65824 + 255) / 256, 256, 0, stream>>>(ws, 65824);
  k_colsum<<<N_PTS / 256, 256, 0, stream>>>(feat, s);
  k_gram<<<dim3(16, 16), 512, 0, stream>>>(feat, G);
  k_mirror<<<IN_F * IN_F / 256, 256, 0, stream>>>(G);
  k_hist<<<N_PTS / 256, 256, 0, stream>>>(cats, counts);
  k_plan<<<1, 1, 0, stream>>>(counts, offc, tileCat, nT);
  k_scatter<<<N_PTS / 256, 256, 0, stream>>>(cats, offc, cursors, perm);
  k_padfill<<<NUM_CAT, 64, 0, stream>>>(counts, offc, perm);
  k_stats<<<CF, 256, 0, stream>>>(W1, G, s, gamma, beta, scale, shift);
  k_gemm2<<<MAX_SUPER, 512, 0, stream>>>(feat, W1, scale, shift,
                                         perm, tileCat, offc, counts, z);
  k_final<<<PADDED_MAX / 8, 256, 0, stream>>>(z, Wc, bias, perm, tileCat,
                                              offc, counts, shifts, slens, out);
}